// LcaModel_24180665877030
// MI455X (gfx1250) — compile-verified
//
#include <hip/hip_runtime.h>

// ---------------------------------------------------------------------------
// LCA (Locally Competitive Algorithm) for MI455X / gfx1250, fp32 WMMA path.
//
//   b = x@w                    (8192x784x2048)
//   g = w^T w - I              (2048x784x2048)
//   99x: u' = u + s*(b - relu(u-l)@g - u)     (8192x2048x2048 each, 98% of FLOPs)
//   a = relu(u-l); recon = a@w^T
//
// Compute-bound (6.8 TFLOP matrix math vs ~30 GB HBM traffic @ 23.3 TB/s).
// All GEMMs run through V_WMMA_F32_16X16X4_F32 (f32 in / f32 accumulate) to
// match the fp32 reference numerics.
//
// v2: double-buffered LDS (72 KB/block, fine vs 320 KB/WGP) with register
// prefetch of the next K-tile issued before the current tile's 64 WMMAs, and
// a single s_barrier per K-step (was 2) — global latency hides behind compute.
//
// Workspace layout (floats): b[8192*2048] | g[2048*2048] | u0[8192*2048] |
// u1[8192*2048]  => 208 MB required in d_ws.
// ---------------------------------------------------------------------------

typedef __attribute__((ext_vector_type(2))) float v2f;
typedef __attribute__((ext_vector_type(8))) float v8f;

#define LCA_B     8192
#define LCA_D     784
#define LCA_L     2048
#define LCA_STEPS 99
#define LCA_S     0.01f
#define LCA_LAM   0.3f

// Block tile geometry
#define BM    128
#define BN    128
#define BKD   32
#define A_LDS 36            // BKD + 4 pad: conflict-free ds_load_b64 A-frags, 16B store align
#define B_ROW 288           // floats per k-pair row: (BN + 16 pad) float2 -> bank-shift 32
#define A_TILE (BM * A_LDS)
#define B_TILE ((BKD / 2) * B_ROW)

// Guarded float4 global load (zero-filled out of bounds).
__device__ __forceinline__ float4 ld4_guard(const float* __restrict__ p, int row, int ld,
                                            int col, int colMax, bool rowOK)
{
    float4 v; v.x = v.y = v.z = v.w = 0.0f;
    if (rowOK) {
        const float* q = p + (size_t)row * (size_t)ld;
        if (col + 3 < colMax) {
            v = *(const float4*)(q + col);
        } else {
            if (col     < colMax) v.x = q[col];
            if (col + 1 < colMax) v.y = q[col + 1];
            if (col + 2 < colMax) v.z = q[col + 2];
            if (col + 3 < colMax) v.w = q[col + 3];
        }
    }
    return v;
}

// ---- staged-tile loaders (global -> regs) and spillers (regs -> LDS) ----

template<bool TRANS_A, bool THRESH_A>
__device__ __forceinline__ void load_a_regs(const float* __restrict__ A, int lda,
                                            int M, int K, int m0, int k0,
                                            int tid, float4 ra[4])
{
    #pragma unroll
    for (int p = 0; p < 4; ++p) {
        const int q = tid + p * 256;
        float4 v;
        if (!TRANS_A) {
            const int m  = q >> 3;             // 0..127
            const int k4 = (q & 7) << 2;       // 0..28
            v = ld4_guard(A, m0 + m, lda, k0 + k4, K, (m0 + m) < M);
        } else {
            const int kk = q >> 5;             // 0..31
            const int m4 = (q & 31) << 2;      // 0..124
            v = ld4_guard(A, k0 + kk, lda, m0 + m4, M, (k0 + kk) < K);
        }
        if (THRESH_A) {
            v.x = fmaxf(v.x - LCA_LAM, 0.0f);
            v.y = fmaxf(v.y - LCA_LAM, 0.0f);
            v.z = fmaxf(v.z - LCA_LAM, 0.0f);
            v.w = fmaxf(v.w - LCA_LAM, 0.0f);
        }
        ra[p] = v;
    }
}

template<bool TRANS_A>
__device__ __forceinline__ void store_a_lds(float* __restrict__ As, int tid, const float4 ra[4])
{
    #pragma unroll
    for (int p = 0; p < 4; ++p) {
        const int q = tid + p * 256;
        if (!TRANS_A) {
            const int m  = q >> 3;
            const int k4 = (q & 7) << 2;
            *(float4*)&As[m * A_LDS + k4] = ra[p];
        } else {
            const int kk = q >> 5;
            const int m4 = (q & 31) << 2;
            As[(m4 + 0) * A_LDS + kk] = ra[p].x;
            As[(m4 + 1) * A_LDS + kk] = ra[p].y;
            As[(m4 + 2) * A_LDS + kk] = ra[p].z;
            As[(m4 + 3) * A_LDS + kk] = ra[p].w;
        }
    }
}

template<bool TRANS_B>
__device__ __forceinline__ void load_b_regs(const float* __restrict__ Bm, int ldb,
                                            int N, int K, int n0, int k0,
                                            int tid, float4 rb[4])
{
    #pragma unroll
    for (int p = 0; p < 4; ++p) {
        const int q = tid + p * 256;
        if (!TRANS_B) {
            const int kk = q >> 5;             // 0..31
            const int n4 = (q & 31) << 2;      // 0..124
            rb[p] = ld4_guard(Bm, k0 + kk, ldb, n0 + n4, N, (k0 + kk) < K);
        } else {
            const int nn = q >> 3;             // 0..127
            const int k4 = (q & 7) << 2;       // even
            rb[p] = ld4_guard(Bm, n0 + nn, ldb, k0 + k4, K, (n0 + nn) < N);
        }
    }
}

template<bool TRANS_B>
__device__ __forceinline__ void store_b_lds(float* __restrict__ Bs, int tid, const float4 rb[4])
{
    #pragma unroll
    for (int p = 0; p < 4; ++p) {
        const int q = tid + p * 256;
        if (!TRANS_B) {
            const int kk = q >> 5;
            const int n4 = (q & 31) << 2;
            const int base = (kk >> 1) * B_ROW + (kk & 1);
            Bs[base + (n4 + 0) * 2] = rb[p].x;
            Bs[base + (n4 + 1) * 2] = rb[p].y;
            Bs[base + (n4 + 2) * 2] = rb[p].z;
            Bs[base + (n4 + 3) * 2] = rb[p].w;
        } else {
            const int nn = q >> 3;
            const int k4 = (q & 7) << 2;       // even
            const int b0 = (k4 >> 1) * B_ROW + nn * 2;
            Bs[b0 + 0]         = rb[p].x;      // k4+0
            Bs[b0 + 1]         = rb[p].y;      // k4+1
            Bs[b0 + B_ROW + 0] = rb[p].z;      // k4+2
            Bs[b0 + B_ROW + 1] = rb[p].w;      // k4+3
        }
    }
}

// C = op_epi( op_a(A) * op_b(B) ), 128x128 block tile, 8 waves of 32.
// TRANS_A:  A[m][k] = Aglob[k*lda + m]   (for w^T w)
// TRANS_B:  B[k][n] = Bglob[n*ldb + k]   (for a @ w^T)
// THRESH_A: A element -> relu(A - lambda)  (LCA 'a' operand computed on the fly)
// EPI: 0 = C=acc   1 = C=acc-I   2 = C = u + s*(b - acc - u)  (reads u_in,b_in)
template<bool TRANS_A, bool TRANS_B, bool THRESH_A, int EPI>
__global__ void __launch_bounds__(256)
gemm_wmma_f32(const float* __restrict__ A, int lda,
              const float* __restrict__ Bm, int ldb,
              float* __restrict__ C, int ldc,
              int M, int N, int K,
              const float* __restrict__ u_in,
              const float* __restrict__ b_in)
{
    __shared__ __align__(16) float As[2][A_TILE];   // [m][k] stride 36, 2 x 18 KB
    __shared__ __align__(16) float Bs[2][B_TILE];   // k-pair interleaved, 2 x 18 KB

    const int tid  = threadIdx.x;
    const int lane = tid & 31;
    const int wv   = tid >> 5;          // wave 0..7
    const int wm   = (wv >> 2) * 64;    // wave M offset in tile (0 / 64)
    const int wn   = (wv & 3) * 32;     // wave N offset in tile (0/32/64/96)
    const int r    = lane & 15;         // row/col within 16-wide frag
    const int h    = lane >> 4;         // half-wave select (K pair / M half)
    const int m0   = blockIdx.y * BM;
    const int n0   = blockIdx.x * BN;

    v8f acc[4][2];
    #pragma unroll
    for (int i = 0; i < 4; ++i)
        #pragma unroll
        for (int j = 0; j < 2; ++j)
            #pragma unroll
            for (int e = 0; e < 8; ++e) acc[i][j][e] = 0.0f;

    float4 ra[4], rb[4];

    // prologue: stage tile 0 into buffer 0
    load_a_regs<TRANS_A, THRESH_A>(A, lda, M, K, m0, 0, tid, ra);
    load_b_regs<TRANS_B>(Bm, ldb, N, K, n0, 0, tid, rb);
    store_a_lds<TRANS_A>(As[0], tid, ra);
    store_b_lds<TRANS_B>(Bs[0], tid, rb);
    __syncthreads();

    const int nk = (K + BKD - 1) / BKD;
    for (int t = 0; t < nk; ++t) {
        const float* __restrict__ curA = As[t & 1];
        const float* __restrict__ curB = Bs[t & 1];

        // issue global loads for next tile first: they retire behind 64 WMMAs
        const bool more = (t + 1 < nk);
        if (more) {
            const int kn = (t + 1) * BKD;
            load_a_regs<TRANS_A, THRESH_A>(A, lda, M, K, m0, kn, tid, ra);
            load_b_regs<TRANS_B>(Bm, ldb, N, K, n0, kn, tid, rb);
        }

        // ----- compute: 8 K-substeps x (4x2) wmma, all lanes active -----
        #pragma unroll
        for (int ks = 0; ks < BKD; ks += 4) {
            const int bk = ks + 2 * h;             // even
            v2f af[4], bf[2];
            #pragma unroll
            for (int i = 0; i < 4; ++i)
                af[i] = *(const v2f*)&curA[(wm + i * 16 + r) * A_LDS + bk];
            #pragma unroll
            for (int j = 0; j < 2; ++j)
                bf[j] = *(const v2f*)&curB[(bk >> 1) * B_ROW + (wn + j * 16 + r) * 2];
            #pragma unroll
            for (int i = 0; i < 4; ++i)
                #pragma unroll
                for (int j = 0; j < 2; ++j)
                    acc[i][j] = __builtin_amdgcn_wmma_f32_16x16x4_f32(
                        false, af[i], false, bf[j], (short)0, acc[i][j], false, false);
        }

        // spill prefetched tile into the other buffer; one barrier per K-step
        if (more) {
            store_a_lds<TRANS_A>(As[(t + 1) & 1], tid, ra);
            store_b_lds<TRANS_B>(Bs[(t + 1) & 1], tid, rb);
        }
        __syncthreads();
    }

    // ----- epilogue: C[m][n] per wmma C-layout (VGPR e: M = 8h+e, N = r) -----
    #pragma unroll
    for (int i = 0; i < 4; ++i) {
        #pragma unroll
        for (int j = 0; j < 2; ++j) {
            const int nCol = n0 + wn + j * 16 + r;
            if (nCol < N) {
                const int mBase = m0 + wm + i * 16 + h * 8;
                #pragma unroll
                for (int e = 0; e < 8; ++e) {
                    const int mRow = mBase + e;
                    if (mRow < M) {
                        const size_t off = (size_t)mRow * (size_t)ldc + nCol;
                        float val = acc[i][j][e];
                        if (EPI == 1) { if (mRow == nCol) val -= 1.0f; }
                        if (EPI == 2) {
                            const float uo = u_in[off];
                            const float bb = b_in[off];
                            val = uo + LCA_S * (bb - val - uo);
                        }
                        C[off] = val;
                    }
                }
            }
        }
    }
}

__global__ void fill_zero_f32(float* __restrict__ p, long n)
{
    long i = (long)blockIdx.x * blockDim.x + threadIdx.x;
    const long stride = (long)gridDim.x * blockDim.x;
    for (; i < n; i += stride) p[i] = 0.0f;
}

__global__ void lca_write_a(const float* __restrict__ u, float* __restrict__ a, long n)
{
    long i = (long)blockIdx.x * blockDim.x + threadIdx.x;
    const long stride = (long)gridDim.x * blockDim.x;
    for (; i < n; i += stride) a[i] = fmaxf(u[i] - LCA_LAM, 0.0f);
}

extern "C" void kernel_launch(void* const* d_in, const int* in_sizes, int n_in,
                              void* d_out, int out_size, void* d_ws, size_t ws_size,
                              hipStream_t stream)
{
    const float* x = (const float*)d_in[0];   // [8192, 784]
    const float* w = (const float*)d_in[1];   // [784, 2048]

    const size_t BL = (size_t)LCA_B * LCA_L;  // 16.8M
    const size_t LL = (size_t)LCA_L * LCA_L;  // 4.2M

    float* ws = (float*)d_ws;                 // needs 208 MB
    float* b  = ws;
    float* g  = b + BL;
    float* u0 = g + LL;
    float* u1 = u0 + BL;

    float* recon = (float*)d_out;                        // [8192, 784]
    float* a_out = recon + (size_t)LCA_B * LCA_D;        // [8192, 2048]

    const dim3 blk(256);

    // b = x @ w
    gemm_wmma_f32<false, false, false, 0>
        <<<dim3(LCA_L / BN, LCA_B / BM), blk, 0, stream>>>(
            x, LCA_D, w, LCA_L, b, LCA_L, LCA_B, LCA_L, LCA_D, nullptr, nullptr);

    // g = w^T w - I
    gemm_wmma_f32<true, false, false, 1>
        <<<dim3(LCA_L / BN, LCA_L / BM), blk, 0, stream>>>(
            w, LCA_L, w, LCA_L, g, LCA_L, LCA_L, LCA_L, LCA_D, nullptr, nullptr);

    // u0 = 0  (d_ws is poisoned by the harness)
    fill_zero_f32<<<dim3(2048), blk, 0, stream>>>(u0, (long)BL);

    // 99 LCA updates, double-buffered u (each launch: read uin, write uout)
    float* uin  = u0;
    float* uout = u1;
    for (int it = 0; it < LCA_STEPS; ++it) {
        gemm_wmma_f32<false, false, true, 2>
            <<<dim3(LCA_L / BN, LCA_B / BM), blk, 0, stream>>>(
                uin, LCA_L, g, LCA_L, uout, LCA_L, LCA_B, LCA_L, LCA_L, uin, b);
        float* t = uin; uin = uout; uout = t;
    }

    // a = relu(u - lambda)  -> second output slot
    lca_write_a<<<dim3(2048), blk, 0, stream>>>(uin, a_out, (long)BL);

    // recon = a @ w^T -> first output slot
    gemm_wmma_f32<false, true, false, 0>
        <<<dim3((LCA_D + BN - 1) / BN, LCA_B / BM), blk, 0, stream>>>(
            a_out, LCA_L, w, LCA_L, recon, LCA_D, LCA_B, LCA_D, LCA_L, nullptr, nullptr);
}